// TaskAdaptiveRouter_29489245454607
// MI455X (gfx1250) — compile-verified
//
#include <hip/hip_runtime.h>
#include <math.h>
#include <stdint.h>

typedef __attribute__((ext_vector_type(2))) float v2f;
typedef __attribute__((ext_vector_type(8))) float v8f;

#define B_  4
#define L_  4096
#define D_  2048
#define DT_ 16
#define E_  64
#define M_  (B_ * L_)          // 16384 tokens
#define WRS (D_ + DT_)         // 2064 W_router row stride

#define TM  64                 // tokens per block
#define KC  64                 // K chunk staged in LDS
#define XS  68                 // x tile LDS stride (floats, 16B-aligned rows)
#define WSD 68                 // W tile LDS stride
#define LGS 65                 // logits LDS stride (conflict-free column reads)
#define NBLK (M_ / TM)         // 256
#define NCH  (D_ / KC)         // 32 K-chunks

// ---- CDNA5 async global->LDS copy (ASYNCcnt-tracked, no VGPR round-trip) ---
__device__ __forceinline__ void async_cp16(const float* g, float* l) {
  unsigned lds = (unsigned)(uintptr_t)l;   // generic ptr low 32 bits = LDS offset
  asm volatile("global_load_async_to_lds_b128 %0, %1, off"
               :: "v"(lds), "v"(g) : "memory");
}

#if __has_builtin(__builtin_amdgcn_s_wait_asynccnt)
#define ASYNC_WAIT(n) __builtin_amdgcn_s_wait_asynccnt(n)
#else
#define ASYNC_WAIT(n) asm volatile("s_wait_asynccnt %0" :: "i"(n) : "memory")
#endif

// ---------------- Kernel 1: task projection + per-(b,e) task logit ----------
__global__ __launch_bounds__(64) void task_kernel(
    const float* __restrict__ z, const float* __restrict__ Wtp,
    const float* __restrict__ btp, const float* __restrict__ Wr,
    const float* __restrict__ ebias, float* __restrict__ tasklogit)
{
  __shared__ float tp[B_][DT_];
  int tid = threadIdx.x;            // 64 threads
  int b = tid >> 4, t = tid & 15;   // (b, t) pairs
  float v = btp[t];
#pragma unroll
  for (int j = 0; j < DT_; ++j) v += z[b * DT_ + j] * Wtp[t * DT_ + j];
  // exact GELU: 0.5*v*(1+erf(v/sqrt(2)))
  tp[b][t] = 0.5f * v * (1.0f + erff(v * 0.70710678118654752f));
  __syncthreads();
  int e = tid;                      // expert per thread
  for (int bb = 0; bb < B_; ++bb) {
    float a = ebias[e];
#pragma unroll
    for (int tt = 0; tt < DT_; ++tt)
      a += tp[bb][tt] * Wr[(size_t)e * WRS + D_ + tt];
    tasklogit[bb * E_ + e] = a;
  }
}

// ---------------- Kernel 2: main router GEMM + softmax + top-2 --------------
__global__ __launch_bounds__(128) void router_main(
    const float* __restrict__ x, const float* __restrict__ Wr,
    const float* __restrict__ tasklogit,
    float* __restrict__ topw, float* __restrict__ topi,
    float* __restrict__ probs,
    float* __restrict__ Ppart, float* __restrict__ Fpart)
{
  __shared__ float xs[2][TM * XS];     // double-buffered x tile; buf0 reused for logits
  __shared__ float wt[2][E_ * WSD];    // double-buffered W tile
  __shared__ int   i1s[TM];
  __shared__ int   i2s[TM];

  const int tid  = threadIdx.x;
  const int wave = tid >> 5;
  const int lane = tid & 31;
  const int m0   = blockIdx.x * TM;    // first token of block
  const int bidx = m0 / L_;            // batch index (uniform in block)

  v8f acc0 = {}, acc1 = {}, acc2 = {}, acc3 = {};

  const int ln16 = lane & 15;
  const int koff = (lane < 16) ? 0 : 2;
  const int arow = wave * 16 + ln16;   // A-matrix row within block tile

  // Per-wave, each chunk issues 8 (x) + 8 (W) async copies -> 16 in flight.
  auto stage = [&](int kc, int buf) {
#pragma unroll
    for (int i = tid; i < TM * KC / 4; i += 128) {
      int row = i >> 4, q = (i & 15) << 2;
      async_cp16(x + (size_t)(m0 + row) * D_ + kc + q, &xs[buf][row * XS + q]);
    }
#pragma unroll
    for (int i = tid; i < E_ * KC / 4; i += 128) {
      int e = i >> 4, q = (i & 15) << 2;
      async_cp16(Wr + (size_t)e * WRS + kc + q, &wt[buf][e * WSD + q]);
    }
  };

  stage(0, 0);                         // prologue: chunk 0 -> buf 0

  for (int ch = 0; ch < NCH; ++ch) {
    const int buf = ch & 1;
    if (ch + 1 < NCH) {
      stage((ch + 1) * KC, buf ^ 1);   // overlap next chunk with compute
      ASYNC_WAIT(16);                  // chunk ch's 16 copies drained (in order)
    } else {
      ASYNC_WAIT(0);
    }
    __syncthreads();                   // all waves' copies for chunk ch complete

    const float* xb = xs[buf];
    const float* wb = wt[buf];
#pragma unroll 4
    for (int k0 = 0; k0 < KC; k0 += 4) {
      // A fragment: lane<16 -> K = k0,k0+1 ; lane>=16 -> K = k0+2,k0+3
      v2f a  = *(const v2f*)(xb + arow * XS + k0 + koff);
      // B fragments (N-per-lane): B[k][n] = Wr[n][k]
      v2f b0 = *(const v2f*)(wb + (ln16 +  0) * WSD + k0 + koff);
      v2f b1 = *(const v2f*)(wb + (ln16 + 16) * WSD + k0 + koff);
      v2f b2 = *(const v2f*)(wb + (ln16 + 32) * WSD + k0 + koff);
      v2f b3 = *(const v2f*)(wb + (ln16 + 48) * WSD + k0 + koff);
      acc0 = __builtin_amdgcn_wmma_f32_16x16x4_f32(false, a, false, b0, (short)0, acc0, false, false);
      acc1 = __builtin_amdgcn_wmma_f32_16x16x4_f32(false, a, false, b1, (short)0, acc1, false, false);
      acc2 = __builtin_amdgcn_wmma_f32_16x16x4_f32(false, a, false, b2, (short)0, acc2, false, false);
      acc3 = __builtin_amdgcn_wmma_f32_16x16x4_f32(false, a, false, b3, (short)0, acc3, false, false);
    }
    __syncthreads();                   // reads done before buffer is overwritten
  }

  // Dump accumulators to LDS (reuse xs[0]): C layout -> VGPR r holds M=r
  // (lanes 0-15) and M=r+8 (lanes 16-31), N = lane%16 within each 16-col tile.
  float* lgbase = xs[0];
  {
    int rbase = wave * 16 + ((lane < 16) ? 0 : 8);
    int c = ln16;
#pragma unroll
    for (int r = 0; r < 8; ++r) {
      lgbase[(rbase + r) * LGS + c +  0] = acc0[r];
      lgbase[(rbase + r) * LGS + c + 16] = acc1[r];
      lgbase[(rbase + r) * LGS + c + 32] = acc2[r];
      lgbase[(rbase + r) * LGS + c + 48] = acc3[r];
    }
  }
  __syncthreads();

  // One thread per token: add task logit, softmax, top-2.
  if (tid < TM) {
    const int row = tid;
    const int m = m0 + row;
    const float* tl = tasklogit + bidx * E_;
    float* lg = lgbase + row * LGS;
    float mx = -3.4e38f;
#pragma unroll 8
    for (int e = 0; e < E_; ++e) {
      float t = lg[e] + tl[e];
      lg[e] = t;
      mx = fmaxf(mx, t);
    }
    float s = 0.f;
#pragma unroll 8
    for (int e = 0; e < E_; ++e) {
      float p = expf(lg[e] - mx);
      lg[e] = p;
      s += p;
    }
    float inv = 1.0f / s;
    float t1 = -1.f, t2 = -1.f;
    int   j1 = 0,    j2 = 0;
#pragma unroll 8
    for (int e = 0; e < E_; ++e) {
      float p = lg[e] * inv;
      lg[e] = p;                                 // store normalized prob
      if (p > t1)      { t2 = t1; j2 = j1; t1 = p; j1 = e; }
      else if (p > t2) { t2 = p;  j2 = e; }
    }
    float denom = t1 + t2 + 1e-8f;
    topw[m * 2 + 0] = t1 / denom;
    topw[m * 2 + 1] = t2 / denom;
    topi[m * 2 + 0] = (float)j1;
    topi[m * 2 + 1] = (float)j2;
    i1s[row] = j1;
    i2s[row] = j2;
  }
  __syncthreads();

  // Coalesced probs store.
  for (int i = tid; i < TM * E_; i += 128) {
    int row = i >> 6, c = i & 63;
    probs[(size_t)m0 * E_ + i] = lgbase[row * LGS + c];
  }

  // Per-block per-expert partials in fixed order (deterministic, no atomics).
  if (tid < E_) {
    const int e = tid;
    float P = 0.f, F = 0.f;
    for (int row = 0; row < TM; ++row) {
      P += lgbase[row * LGS + e];
      F += (i1s[row] == e) ? 0.5f : 0.f;
      F += (i2s[row] == e) ? 0.5f : 0.f;
    }
    Ppart[blockIdx.x * E_ + e] = P;
    Fpart[blockIdx.x * E_ + e] = F;
  }
}

// ---------------- Kernel 3: aux loss reduction ------------------------------
__global__ __launch_bounds__(64) void aux_kernel(
    const float* __restrict__ Ppart, const float* __restrict__ Fpart,
    float* __restrict__ out)
{
  __shared__ float red[E_];
  int e = threadIdx.x;
  float P = 0.f, F = 0.f;
  for (int b = 0; b < NBLK; ++b) {
    P += Ppart[b * E_ + e];
    F += Fpart[b * E_ + e];
  }
  const float invM = 1.0f / (float)M_;
  red[e] = (F * invM) * (P * invM);
  __syncthreads();
  for (int s = 32; s > 0; s >>= 1) {
    if (e < s) red[e] += red[e + s];
    __syncthreads();
  }
  if (e == 0) out[0] = (float)E_ * red[0];
}

// ---------------- Launch ----------------------------------------------------
extern "C" void kernel_launch(void* const* d_in, const int* in_sizes, int n_in,
                              void* d_out, int out_size, void* d_ws, size_t ws_size,
                              hipStream_t stream)
{
  const float* x     = (const float*)d_in[0];
  const float* z     = (const float*)d_in[1];
  const float* Wr    = (const float*)d_in[2];
  const float* Wtp   = (const float*)d_in[3];
  const float* btp   = (const float*)d_in[4];
  const float* ebias = (const float*)d_in[5];

  float* out   = (float*)d_out;
  float* topw  = out;                       // [B,L,2]  = 32768
  float* topi  = out + 2 * M_;              // [B,L,2]  = 32768
  float* probs = out + 4 * M_;              // [B,L,E]  = 1048576
  float* aux   = out + 4 * M_ + M_ * E_;    // scalar

  float* ws        = (float*)d_ws;
  float* tasklogit = ws;                    // 256 floats
  float* Ppart     = ws + 256;              // NBLK*64
  float* Fpart     = Ppart + NBLK * E_;     // NBLK*64

  task_kernel<<<1, 64, 0, stream>>>(z, Wtp, btp, Wr, ebias, tasklogit);
  router_main<<<NBLK, 128, 0, stream>>>(x, Wr, tasklogit, topw, topi, probs,
                                        Ppart, Fpart);
  aux_kernel<<<1, 64, 0, stream>>>(Ppart, Fpart, aux);
}